// PathRewardWithCollision_62328565399611
// MI455X (gfx1250) — compile-verified
//
#include <hip/hip_runtime.h>
#include <stdint.h>

// Problem shape (fixed by reference setup_inputs).
constexpr int M = 16;   // modes
constexpr int T = 80;   // timesteps
constexpr int O = 64;   // other agents
constexpr float K_OFFSET = 8.0f;

// ---- CDNA5 async global->LDS copy (gfx1250 GLOBAL_LOAD_ASYNC_TO_LDS_B128) ----
typedef __attribute__((address_space(3))) void lds_void_t;

__device__ __forceinline__ void async_copy_b128(void* lds_dst, const void* gsrc) {
    // LDS byte offset (32-bit) via addrspacecast generic -> AS3.
    unsigned int lds_off =
        (unsigned int)(unsigned long long)(lds_void_t*)lds_dst;
    unsigned long long gaddr = (unsigned long long)(uintptr_t)gsrc;
    asm volatile("global_load_async_to_lds_b128 %0, %1, off"
                 :: "v"(lds_off), "v"(gaddr)
                 : "memory");
}

__device__ __forceinline__ void wait_asynccnt0() {
    asm volatile("s_wait_asynccnt 0" ::: "memory");
}

// One workgroup per agent `a`. 256 threads = 8 wave32.
// Thread tid -> mode m = tid>>4, timestep subgroup t0 = tid&15 (5 t's each).
__global__ __launch_bounds__(256) void path_reward_collision_kernel(
    const float* __restrict__ targets,          // (A,T,2)
    const int*   __restrict__ target_mask,      // (A,T)
    const float* __restrict__ traj_preds,       // (A,M,T,4)
    const float* __restrict__ agent_fut_width,  // (A,T)
    const float* __restrict__ other_fut_trajs,  // (A,O,T,2)
    const int*   __restrict__ other_fut_masks,  // (A,O,T)
    const float* __restrict__ other_fut_widths, // (A,O,T)
    float*       __restrict__ out,              // 3 x (A,M): total, coll_pen, l2_pen
    int A)
{
    __shared__ __align__(16) float s_xy[O * T * 2];  // raw other trajs (x,y)   40 KB
    __shared__ __align__(16) float s_tp[M * T * 4];  // raw traj_preds tile     20 KB
    __shared__ float s_hwo[O * T];                   // 0.5*w_o if mask else -1e30  20 KB
    __shared__ float s_tx[T], s_ty[T], s_hwi[T];
    __shared__ int   s_tm[T];
    __shared__ float s_l2[M], s_coll[M];

    const int a   = blockIdx.x;
    const int tid = threadIdx.x;

    // --- Stage other_fut_trajs[a] (O*T*2 floats = 40960 B) via async DMA to LDS ---
    const float* g_ot = other_fut_trajs + (size_t)a * (O * T * 2);
    #pragma unroll
    for (int i = 0; i < (O * T * 2) / (256 * 4); ++i) {   // 10 iters x 16B/thread
        const int idx = (i * 256 + tid) * 4;              // float index, 16B aligned
        async_copy_b128(&s_xy[idx], g_ot + idx);
    }

    // --- Stage traj_preds[a] (M*T*4 floats = 20480 B) via async DMA to LDS ---
    const float* g_tp = traj_preds + (size_t)a * (M * T * 4);
    #pragma unroll
    for (int i = 0; i < (M * T * 4) / (256 * 4); ++i) {   // 5 iters x 16B/thread
        const int idx = (i * 256 + tid) * 4;
        async_copy_b128(&s_tp[idx], g_tp + idx);
    }

    // --- Fuse masks+widths into one LDS array while the DMAs fly ---
    const int*   g_om = other_fut_masks  + (size_t)a * (O * T);
    const float* g_ow = other_fut_widths + (size_t)a * (O * T);
    for (int i = tid; i < O * T; i += 256) {
        s_hwo[i] = (g_om[i] > 0) ? 0.5f * g_ow[i] : -1.0e30f;
    }

    if (tid < T) {
        s_tm[tid]  = target_mask[(size_t)a * T + tid];
        s_tx[tid]  = targets[((size_t)a * T + tid) * 2 + 0];
        s_ty[tid]  = targets[((size_t)a * T + tid) * 2 + 1];
        s_hwi[tid] = 0.5f * agent_fut_width[(size_t)a * T + tid];
    }
    if (tid < M) { s_l2[tid] = 0.0f; s_coll[tid] = 0.0f; }

    wait_asynccnt0();   // this wave's async LDS writes have landed
    __syncthreads();    // all waves' stages visible

    const int m  = tid >> 4;   // 0..15
    const int t0 = tid & 15;   // 0..15
    const float2* s_xy2 = reinterpret_cast<const float2*>(s_xy);
    const float2* s_tp2 = reinterpret_cast<const float2*>(s_tp);  // (m*T+t)*2 -> (x,y)

    float l2acc = 0.0f, collacc = 0.0f;
    #pragma unroll
    for (int ti = 0; ti < T / 16; ++ti) {
        const int t = t0 + ti * 16;
        if (s_tm[t] > 0) {                 // tm==0: zero l2 term and valid=false
            const float2 pxy = s_tp2[(m * T + t) * 2];   // (x,y) of pred, ds_load_b64
            const float px = pxy.x;
            const float py = pxy.y;
            const float dx = px - s_tx[t];
            const float dy = py - s_ty[t];
            l2acc += sqrtf(dx * dx + dy * dy);

            const float hwi = s_hwi[t];
            int hit = 0;
            #pragma unroll 8
            for (int o = 0; o < O; ++o) {
                const float2 oxy = s_xy2[o * T + t];       // ds_load_b64
                const float  rad = hwi + s_hwo[o * T + t]; // ~-1e30 when invalid
                const float ddx = px - oxy.x;
                const float ddy = py - oxy.y;
                const float dist = sqrtf(ddx * ddx + ddy * ddy);
                hit |= (dist <= rad) ? 1 : 0;              // matches reference compare
            }
            collacc += (float)hit;                          // any over O, summed over t
        }
    }

    atomicAdd(&s_l2[m],   l2acc);    // ds_add_f32
    atomicAdd(&s_coll[m], collacc);
    __syncthreads();

    if (tid < M) {
        const float total_l2 = s_l2[tid];
        const float coll     = s_coll[tid];
        const float l2p = -(total_l2 - K_OFFSET);
        const float cp  = -coll * 100.0f;
        const size_t AM   = (size_t)A * M;
        const size_t base = (size_t)a * M + tid;
        out[base]          = l2p + cp;  // total
        out[AM + base]     = cp;        // collision_penalty
        out[2 * AM + base] = l2p;       // l2_penalty
    }
}

extern "C" void kernel_launch(void* const* d_in, const int* in_sizes, int n_in,
                              void* d_out, int out_size, void* d_ws, size_t ws_size,
                              hipStream_t stream) {
    const float* targets          = (const float*)d_in[0];
    const int*   target_mask      = (const int*)  d_in[1];
    const float* traj_preds       = (const float*)d_in[2];
    // d_in[3] = mode_probs: unused by the reference computation.
    const float* agent_fut_width  = (const float*)d_in[4];
    const float* other_fut_trajs  = (const float*)d_in[5];
    const int*   other_fut_masks  = (const int*)  d_in[6];
    const float* other_fut_widths = (const float*)d_in[7];
    float* out = (float*)d_out;

    const int A = in_sizes[1] / T;   // target_mask is (A,T)

    path_reward_collision_kernel<<<A, 256, 0, stream>>>(
        targets, target_mask, traj_preds, agent_fut_width,
        other_fut_trajs, other_fut_masks, other_fut_widths, out, A);
}